// GraphConvolution_48576080118484
// MI455X (gfx1250) — compile-verified
//
#include <hip/hip_runtime.h>

typedef __attribute__((ext_vector_type(2))) float v2f;
typedef __attribute__((ext_vector_type(8))) float v8f;

#define NNODES 50000
#define INDIM 256
#define OUTDIM 64
#define SUPPORT 8
#define NBASES 4
#define MTILES (NNODES / 16)      // 3125, exact
#define MPAIRS ((MTILES + 1) / 2) // 1563 (last pair is a singleton)

// ---------------------------------------------------------------------------
// Phase 1: WT[s][o][i] = sum_b comp[s,b] * W_F[b][i][o]
// Transposed layout so a B-fragment {W[k][n], W[k+1][n]} is one contiguous
// float2 at WT + n*INDIM + k. Writes fully coalesced (i fastest).
// ---------------------------------------------------------------------------
__global__ __launch_bounds__(256) void rgcn_combine(
    const float* __restrict__ WF, const float* __restrict__ comp,
    float* __restrict__ WT) {
  int t = blockIdx.x * blockDim.x + threadIdx.x;
  if (t >= SUPPORT * INDIM * OUTDIM) return;
  int i = t & (INDIM - 1);
  int o = (t >> 8) & (OUTDIM - 1);
  int s = t >> 14;
  float acc = 0.f;
#pragma unroll
  for (int bb = 0; bb < NBASES; ++bb)
    acc += comp[s * NBASES + bb] * WF[bb * (INDIM * OUTDIM) + i * OUTDIM + o];
  WT[t] = acc;
}

// ---------------------------------------------------------------------------
// Phase 2: FW[sLoc] = E @ W[s] via V_WMMA_F32_16X16X4_F32 (wave32).
// Block = 128 threads = 4 waves; each wave owns a PAIR of M-tiles (32 rows)
// x 4 N-tiles = 8 accumulators (64 VGPRs). B fragments (one b64 from the
// 64KB L0/L2-resident WT[s]) are shared across both M-tiles:
// per K-step 6 x b64 loads feed 8 WMMAs.
// A-frag (16x4 f32): lane<16 -> {E[r][k],E[r][k+1]}, lane>=16 -> {k+2,k+3}.
// Last (odd) M-tile pair clamps its load rows and predicates the store
// wave-uniformly, so EXEC is all-1s for every WMMA.
// ---------------------------------------------------------------------------
__global__ __launch_bounds__(128) void rgcn_gemm(
    const float* __restrict__ E, const float* __restrict__ WTg,
    float* __restrict__ FW, int sBase) {
  const int wave = threadIdx.x >> 5;
  const int lane = threadIdx.x & 31;
  const int hi = lane >> 4;
  const int l16 = lane & 15;
  const int pair = blockIdx.x * 4 + wave; // wave-uniform
  if (pair >= MPAIRS) return;             // whole-wave exit
  const int mt0 = pair * 2;
  const int mt1 = mt0 + 1;
  const int mt1c = (mt1 < MTILES) ? mt1 : (MTILES - 1); // clamp for loads
  const bool st1 = (mt1 < MTILES);                      // wave-uniform

  const int sLoc = blockIdx.y;
  const int s = sBase + sLoc;
  const float* WT = WTg + (size_t)s * (INDIM * OUTDIM);
  const float* erow0 = E + (size_t)(mt0 * 16 + l16) * INDIM + 2 * hi;
  const float* erow1 = E + (size_t)(mt1c * 16 + l16) * INDIM + 2 * hi;
  const float* wrow = WT + (size_t)l16 * INDIM + 2 * hi;

  v8f acc[2][4];
#pragma unroll
  for (int m = 0; m < 2; ++m)
#pragma unroll
    for (int i = 0; i < 4; ++i)
#pragma unroll
      for (int j = 0; j < 8; ++j) acc[m][i][j] = 0.0f;

#pragma unroll 4
  for (int k = 0; k < INDIM; k += 4) {
    v2f a0 = *(const v2f*)(erow0 + k);
    v2f a1 = *(const v2f*)(erow1 + k);
#pragma unroll
    for (int nt = 0; nt < 4; ++nt) {
      v2f b = *(const v2f*)(wrow + (size_t)nt * 16 * INDIM + k);
      acc[0][nt] = __builtin_amdgcn_wmma_f32_16x16x4_f32(
          false, a0, false, b, (short)0, acc[0][nt], false, false);
      acc[1][nt] = __builtin_amdgcn_wmma_f32_16x16x4_f32(
          false, a1, false, b, (short)0, acc[1][nt], false, false);
    }
  }

  // C/D 16x16 f32 layout: VGPR r -> (M=r, N=lane) lanes 0-15,
  // (M=r+8, N=lane-16) lanes 16-31.
  float* out0 =
      FW + ((size_t)sLoc * NNODES + mt0 * 16 + 8 * hi) * OUTDIM + l16;
#pragma unroll
  for (int nt = 0; nt < 4; ++nt)
#pragma unroll
    for (int r = 0; r < 8; ++r)
      out0[(size_t)r * OUTDIM + nt * 16] = acc[0][nt][r];

  if (st1) {
    float* out1 =
        FW + ((size_t)sLoc * NNODES + mt1 * 16 + 8 * hi) * OUTDIM + l16;
#pragma unroll
    for (int nt = 0; nt < 4; ++nt)
#pragma unroll
      for (int r = 0; r < 8; ++r)
        out1[(size_t)r * OUTDIM + nt * 16] = acc[1][nt][r];
  }
}

// ---------------------------------------------------------------------------
// Phase 3: msgs = A_vals * FW[A_cols]; out[A_rows] += msgs.
// 64 lanes per nnz -> coalesced gather + coalesced fp32 atomic scatter.
// ---------------------------------------------------------------------------
__global__ __launch_bounds__(256) void rgcn_scatter(
    const float* __restrict__ FW, const float* __restrict__ vals,
    const int* __restrict__ rows, const int* __restrict__ cols,
    float* __restrict__ out, int sBase, int sCount, int nnz) {
  long long t = (long long)blockIdx.x * blockDim.x + threadIdx.x;
  int e = (int)(t >> 6);
  int o = (int)(t & 63);
  if (e >= nnz) return;
  int c = cols[e];
  int s = c / NNODES;
  if (s < sBase || s >= sBase + sCount) return;
  int n = c - s * NNODES;
  float m = vals[e] * FW[((size_t)(s - sBase) * NNODES + n) * OUTDIM + o];
  __hip_atomic_fetch_add(&out[(size_t)rows[e] * OUTDIM + o], m,
                         __ATOMIC_RELAXED, __HIP_MEMORY_SCOPE_AGENT);
}

// ---------------------------------------------------------------------------
// Phase 4: out = relu(out + bias)
// ---------------------------------------------------------------------------
__global__ __launch_bounds__(256) void rgcn_finalize(
    float* __restrict__ out, const float* __restrict__ bias) {
  int t = blockIdx.x * blockDim.x + threadIdx.x;
  if (t >= NNODES * OUTDIM) return;
  float v = out[t] + bias[t & (OUTDIM - 1)];
  out[t] = v > 0.f ? v : 0.f;
}

extern "C" void kernel_launch(void* const* d_in, const int* in_sizes, int n_in,
                              void* d_out, int out_size, void* d_ws,
                              size_t ws_size, hipStream_t stream) {
  const float* E = (const float*)d_in[0];
  const float* WF = (const float*)d_in[1];
  const float* comp = (const float*)d_in[2];
  const float* bias = (const float*)d_in[3];
  const float* Avals = (const float*)d_in[4];
  const int* Arows = (const int*)d_in[5];
  const int* Acols = (const int*)d_in[6];
  float* out = (float*)d_out;
  const int nnz = in_sizes[4];

  float* WT = (float*)d_ws;
  const size_t wBytes = (size_t)SUPPORT * INDIM * OUTDIM * sizeof(float);
  float* FW = (float*)((char*)d_ws + wBytes);
  const size_t fwFull = (size_t)SUPPORT * NNODES * OUTDIM * sizeof(float);

  hipMemsetAsync(d_out, 0, (size_t)NNODES * OUTDIM * sizeof(float), stream);

  {
    const int total = SUPPORT * INDIM * OUTDIM;
    rgcn_combine<<<(total + 255) / 256, 256, 0, stream>>>(WF, comp, WT);
  }

  const int gemmBlocksX = (MPAIRS + 3) / 4;
  const long long scatterThreads = (long long)nnz * 64;
  const int scatterBlocks = (int)((scatterThreads + 255) / 256);

  if (ws_size >= wBytes + fwFull) {
    // FW for all 8 relations fits in workspace (and in the 192MB L2).
    rgcn_gemm<<<dim3(gemmBlocksX, SUPPORT), 128, 0, stream>>>(E, WT, FW, 0);
    rgcn_scatter<<<scatterBlocks, 256, 0, stream>>>(FW, Avals, Arows, Acols,
                                                    out, 0, SUPPORT, nnz);
  } else {
    // Small-workspace fallback: one relation at a time (~13MB),
    // stream order serializes each gemm -> scatter pair.
    for (int s = 0; s < SUPPORT; ++s) {
      rgcn_gemm<<<dim3(gemmBlocksX, 1), 128, 0, stream>>>(E, WT, FW, s);
      rgcn_scatter<<<scatterBlocks, 256, 0, stream>>>(FW, Avals, Arows, Acols,
                                                      out, s, 1, nnz);
    }
  }

  rgcn_finalize<<<((NNODES * OUTDIM) + 255) / 256, 256, 0, stream>>>(out, bias);
}